// Encoder_6760278524058
// MI455X (gfx1250) — compile-verified
//
#include <hip/hip_runtime.h>
#include <hip/hip_bf16.h>

typedef __attribute__((ext_vector_type(2))) float v2f;
typedef __attribute__((ext_vector_type(4))) float v4f;
typedef __attribute__((ext_vector_type(8))) float v8f;

// ---------------------------------------------------------------------------
// Degree accumulation: deg[dst] += 1 per edge (reference: segment_sum(ones,dst))
// ---------------------------------------------------------------------------
__global__ void gcn_degree_kernel(const int* __restrict__ dst,
                                  float* __restrict__ deg, int E) {
    int e = blockIdx.x * blockDim.x + threadIdx.x;
    if (e < E) atomicAdd(&deg[dst[e]], 1.0f);
}

// dinv[i] = rsqrt(deg[i] + 1)   (+1 = self loop), computed in place
__global__ void gcn_dinv_kernel(float* __restrict__ deg, int M) {
    int i = blockIdx.x * blockDim.x + threadIdx.x;
    if (i < M) deg[i] = rsqrtf(deg[i] + 1.0f);
}

// ---------------------------------------------------------------------------
// H[M,N] = X[M,K] @ W[K,N] with V_WMMA_F32_16X16X4_F32 (fp32 in, fp32 acc).
// One 16-row strip of X per block staged in LDS (float4 loads); wave w owns
// N-tile [16w,16w+16).  ISA layouts (CDNA5 ISA §7.12.2):
//   A 16x4 : lanes 0-15 -> M=lane, VGPR{0,1} = K{0,1}; lanes 16-31 -> K{2,3}
//   B 4x16 : lanes hold N=lane&15; lane-half selects K pair (mirrors A)
//   C 16x16: VGPR v -> M = v + 8*(lane>=16), N = lane&15
// ---------------------------------------------------------------------------
template <int K, int N>
__global__ __launch_bounds__((N / 16) * 32) void gcn_gemm_wmma(
    const float* __restrict__ X, const float* __restrict__ W,
    float* __restrict__ H, int M) {
    __shared__ float As[16 * K];
    const int mbase    = blockIdx.x * 16;
    const int tid      = threadIdx.x;
    const int nthreads = (N / 16) * 32;
    const bool full    = (mbase + 16 <= M);

    // Cooperative float4 staging of the 16 x K fp32 A-tile (zero-pad past M).
    constexpr int KV = K / 4;                // float4s per row
    const v4f* X4 = reinterpret_cast<const v4f*>(X);
    v4f* As4      = reinterpret_cast<v4f*>(As);
    for (int idx = tid; idx < 16 * KV; idx += nthreads) {
        int r = idx / KV;
        int row = mbase + r;
        v4f v = {};
        if (full || row < M) v = X4[(size_t)row * KV + (idx - r * KV)];
        As4[idx] = v;
    }
    __syncthreads();

    const int lane  = tid & 31;
    const int wave  = tid >> 5;
    const int col   = wave * 16 + (lane & 15); // this lane's N column
    const int khalf = lane >> 4;               // 0 -> K{0,1}, 1 -> K{2,3}

    v8f acc = {};
#pragma unroll 4
    for (int k = 0; k < K; k += 4) {
        const int ka = k + khalf * 2;
        v2f a = *reinterpret_cast<const v2f*>(&As[(lane & 15) * K + ka]);
        v2f b;
        b.x = W[(size_t)ka * N + col];
        b.y = W[(size_t)(ka + 1) * N + col];
        acc = __builtin_amdgcn_wmma_f32_16x16x4_f32(
            /*neg_a=*/false, a, /*neg_b=*/false, b,
            /*c_mod=*/(short)0, acc, /*reuse_a=*/false, /*reuse_b=*/false);
    }

    const int rbase = mbase + khalf * 8;
    float* Hp = H + (size_t)rbase * N + col;
    if (full) {
        // Uniform fast path: 8 straight stores, immediate offsets, clauseable.
#pragma unroll
        for (int v = 0; v < 8; ++v) Hp[(size_t)v * N] = acc[v];
    } else {
#pragma unroll
        for (int v = 0; v < 8; ++v)
            if (rbase + v < M) Hp[(size_t)v * N] = acc[v];
    }
}

// ---------------------------------------------------------------------------
// Edge scatter: one wave per edge.  agg[dst] += h[src] * dinv[src]*dinv[dst].
// Edge id is forced into an SGPR (readfirstlane) so src/dst/dinv lower to
// scalar (SMEM) loads; gather of h[src] is a single b128/b64 per lane.
// agg (<=51MB) is L2-resident on MI455X (192MB L2) -> atomics stay on chip.
// ---------------------------------------------------------------------------
template <int N>
__global__ void gcn_scatter(const float* __restrict__ h,
                            const int* __restrict__ src,
                            const int* __restrict__ dst,
                            const float* __restrict__ dinv,
                            float* __restrict__ agg, int E) {
    long long gtid = (long long)blockIdx.x * blockDim.x + threadIdx.x;
    int e    = __builtin_amdgcn_readfirstlane((int)(gtid >> 5)); // wave-uniform
    int lane = threadIdx.x & 31;
    if (e >= E) return;
    int s = src[e];
    int d = dst[e];
    float norm = dinv[s] * dinv[d];
    const float* hs = h + (size_t)s * N;
    float* ad       = agg + (size_t)d * N;
    constexpr int V = N / 32;                 // floats per lane (4 or 2)
    v4f hv = {};
    if (V == 4) {
        hv = reinterpret_cast<const v4f*>(hs)[lane];
    } else {
        v2f t = reinterpret_cast<const v2f*>(hs)[lane];
        hv.x = t.x; hv.y = t.y;
    }
#pragma unroll
    for (int j = 0; j < V; ++j)
        atomicAdd(&ad[lane * V + j], hv[j] * norm);
}

// ---------------------------------------------------------------------------
// out = relu(agg + h * dinv^2 + bias)  (in-place over agg is safe: 1 rd, 1 wr)
// ---------------------------------------------------------------------------
template <int N>
__global__ void gcn_finalize(const float* __restrict__ agg,
                             const float* __restrict__ h,
                             const float* __restrict__ dinv,
                             const float* __restrict__ bias,
                             float* __restrict__ out, int M) {
    long long i = (long long)blockIdx.x * blockDim.x + threadIdx.x;
    if (i >= (long long)M * N) return;
    int node = (int)(i / N);
    int c    = (int)(i - (long long)node * N);
    float di = dinv[node];
    float v  = agg[i] + h[i] * di * di + bias[c];
    out[i] = fmaxf(v, 0.0f);
}

extern "C" void kernel_launch(void* const* d_in, const int* in_sizes, int n_in,
                              void* d_out, int out_size, void* d_ws,
                              size_t ws_size, hipStream_t stream) {
    const float* x  = (const float*)d_in[0];
    const float* W1 = (const float*)d_in[1];
    const float* b1 = (const float*)d_in[2];
    const float* W2 = (const float*)d_in[3];
    const float* b2 = (const float*)d_in[4];
    const int*   ei = (const int*)d_in[5];   // [2, E] int32 row-major

    constexpr int IN = 256, H1 = 128, H2 = 64;
    const int M = in_sizes[0] / IN;          // 100000 (multiple of 16)
    const int E = in_sizes[5] / 2;           // 1600000
    const int* src = ei;
    const int* dst = ei + E;

    // Workspace layout (M*(128+128+1)*4 ~= 103 MB):
    float* h1   = (float*)d_ws;              // [M,128]; reused as h2 later
    float* agg1 = h1 + (size_t)M * H1;       // [M,128]; becomes out1 in place
    float* dinv = agg1 + (size_t)M * H1;     // [M]; deg then rsqrt in place
    float* h2   = h1;                        // h1 dead after finalize-1
    float* out  = (float*)d_out;             // doubles as agg2

    // Scratch must be (re)zeroed every call (harness poisons once only).
    hipMemsetAsync(dinv, 0, (size_t)M * sizeof(float), stream);
    hipMemsetAsync(agg1, 0, (size_t)M * H1 * sizeof(float), stream);
    hipMemsetAsync(out, 0, (size_t)M * H2 * sizeof(float), stream);

    gcn_degree_kernel<<<(E + 255) / 256, 256, 0, stream>>>(dst, dinv, E);
    gcn_dinv_kernel<<<(M + 255) / 256, 256, 0, stream>>>(dinv, M);

    const long long sthreads = (long long)E * 32;
    const int sblocks = (int)((sthreads + 255) / 256);

    // ---- Layer 1: h1 = x @ W1, aggregate, relu(+b1) ----
    gcn_gemm_wmma<IN, H1>
        <<<(M + 15) / 16, (H1 / 16) * 32, 0, stream>>>(x, W1, h1, M);
    gcn_scatter<H1><<<sblocks, 256, 0, stream>>>(h1, src, dst, dinv, agg1, E);
    gcn_finalize<H1>
        <<<(int)(((long long)M * H1 + 255) / 256), 256, 0, stream>>>(
            agg1, h1, dinv, b1, agg1, M);          // out1 lives in agg1

    // ---- Layer 2: h2 = out1 @ W2, aggregate into d_out, relu(+b2) ----
    gcn_gemm_wmma<H1, H2>
        <<<(M + 15) / 16, (H2 / 16) * 32, 0, stream>>>(agg1, W2, h2, M);
    gcn_scatter<H2><<<sblocks, 256, 0, stream>>>(h2, src, dst, dinv, out, E);
    gcn_finalize<H2>
        <<<(int)(((long long)M * H2 + 255) / 256), 256, 0, stream>>>(
            out, h2, dinv, b2, out, M);
}